// QuantumLSTM_65481071395225
// MI455X (gfx1250) — compile-verified
//
#include <hip/hip_runtime.h>
#include <math.h>

// ---------------------------------------------------------------------------
// Quantum LSTM, algebraically reduced:
//   The 2^10-amplitude circuit collapses exactly to prefix products of
//   cos(z + theta):  E[Z_k] = prod_{j=0..k} C_j (k>=1), E[Z_0] = prod_{j=1..9} C_j.
// Phase 1 (parallel, WMMA f32 16x16x4): pre = X @ Wx^T + (b + theta)
// Phase 2 (sequential scan, 1 thread per batch lane, hx/cx in registers).
// ---------------------------------------------------------------------------

typedef __attribute__((ext_vector_type(2))) float v2f;
typedef __attribute__((ext_vector_type(8))) float v8f;

#define TT 64
#define BB 256
#define DD 246
#define HH 10
#define NO 40      // 4 gates * 10 wires
#define FAN 256    // D + H

// ---------------------------------------------------------------------------
// Phase 1: pre = X @ Wx^T + (b + theta), FP32 WMMA 16x16x4.
// M = 16384, N = 40 (pad to 48 -> 3 N-tiles), K = 246.
// One wave per 16x16 output tile; 4 waves / 128-thread block.
// A (16x4 f32): lanes 0-15 hold {K=k,k+1}, lanes 16-31 hold {K=k+2,k+3}.
// B (4x16 f32): N = lane&15; same K split by half-wave.
// C/D v8f: VGPR j -> M=j (lanes 0-15) / M=j+8 (lanes 16-31), N = lane&15.
// Main loop k=0..240 is completely unguarded:
//   - all A rows exist (M exact multiple of 16)
//   - padded B columns (col>=40) read row 0 harmlessly: WMMA output columns
//     are independent and we never store columns >= 40.
// Tail (K=244,245): one WMMA; hi half-wave zeroed via selects (no branches),
//   loads clamped to K=244 so the last x row never reads out of bounds.
// ---------------------------------------------------------------------------
__global__ __launch_bounds__(128) void qlstm_preact_wmma(
    const float* __restrict__ x,      // (T,B,246)
    const float* __restrict__ W,      // (4,10,256) -> 40 rows x 256
    const float* __restrict__ bvec,   // (4,10) -> 40
    const float* __restrict__ theta,  // (4,10) -> 40
    float* __restrict__ pre)          // (T*B, 40)
{
    const int wid   = (blockIdx.x * blockDim.x + threadIdx.x) >> 5; // 0..3071
    const int lane  = threadIdx.x & 31;
    const int mTile = wid / 3;
    const int nTile = wid % 3;
    const int m0 = mTile * 16;
    const int n0 = nTile * 16;

    const bool hi   = lane >= 16;
    const int laneM = lane & 15;          // A row within tile; also N column
    const int kOff  = hi ? 2 : 0;

    const int row    = m0 + laneM;            // global x row (< 16384 always)
    const int col    = n0 + laneM;            // output column (g*10+o)
    const bool colOk = col < NO;

    const float* ap = x + (size_t)row * DD + kOff;
    const float* bp = W + (size_t)(colOk ? col : 0) * FAN + kOff;

    v8f acc = {};
    // K = 0..243: fully in-bounds for both half-waves, zero guards.
    #pragma unroll 4
    for (int k = 0; k < 244; k += 4) {
        v2f a  = *(const v2f*)ap;
        v2f bm = *(const v2f*)bp;
        acc = __builtin_amdgcn_wmma_f32_16x16x4_f32(
                  false, a, false, bm, (short)0, acc, false, false);
        ap += 4;
        bp += 4;
    }
    // K tail = {244,245}; hi half-wave (K=246,247) contributes zero.
    {
        const float* at = x + (size_t)row * DD + 244;  // clamped: always in-bounds
        const float* bt = W + (size_t)(colOk ? col : 0) * FAN + 244;
        v2f a  = *(const v2f*)at;
        v2f bm = *(const v2f*)bt;
        a.x  = hi ? 0.0f : a.x;   // selects (v_cndmask), not branches
        a.y  = hi ? 0.0f : a.y;
        bm.x = hi ? 0.0f : bm.x;
        bm.y = hi ? 0.0f : bm.y;
        acc = __builtin_amdgcn_wmma_f32_16x16x4_f32(
                  false, a, false, bm, (short)0, acc, false, false);
    }

    if (colOk) {
        const float bias = bvec[col] + theta[col];  // fold theta: cos(z + theta)
        const int rbase = m0 + (hi ? 8 : 0);
        #pragma unroll
        for (int j = 0; j < 8; ++j)
            pre[(rbase + j) * NO + col] = acc[j] + bias;
    }
}

// ---------------------------------------------------------------------------
// Phase 2: sequential scan over T. One thread per batch element; hx/cx in
// registers; Wh (40x10) staged in LDS (broadcast reads, conflict-free).
// ---------------------------------------------------------------------------
__device__ __forceinline__ float sigmoidf(float v) {
    return 1.0f / (1.0f + expf(-v));
}

__global__ __launch_bounds__(32) void qlstm_recurrence(
    const float* __restrict__ pre,   // (T*B, 40), bias+theta already folded
    const float* __restrict__ W,     // (40, 256); columns 246..255 are Wh
    float* __restrict__ out)         // outputs(T,B,10) ++ hx(B,10) ++ cx(B,10)
{
    __shared__ float Wh[NO * HH];
    for (int i = threadIdx.x; i < NO * HH; i += 32) {
        const int n = i / HH, h = i % HH;
        Wh[i] = W[n * FAN + DD + h];
    }
    __syncthreads();

    const int b = blockIdx.x * 32 + threadIdx.x;   // 8 blocks * 32 = 256

    float hx[HH], cx[HH];
    #pragma unroll
    for (int w = 0; w < HH; ++w) { hx[w] = 0.0f; cx[w] = 0.0f; }

    for (int t = 0; t < TT; ++t) {
        const float* p = pre + (t * BB + b) * NO;

        float arg[NO];
        #pragma unroll
        for (int n = 0; n < NO; ++n) {
            float s = p[n];
            #pragma unroll
            for (int h = 0; h < HH; ++h) s = fmaf(hx[h], Wh[n * HH + h], s);
            arg[n] = s;
        }

        // Quantum layer, closed form: prefix products of cos(arg).
        float E[4][HH];
        #pragma unroll
        for (int g = 0; g < 4; ++g) {
            float C[HH];
            #pragma unroll
            for (int w = 0; w < HH; ++w) C[w] = cosf(arg[g * HH + w]);
            float run  = C[0];   // prod_{0..k}
            float tail = 1.0f;   // prod_{1..9}
            #pragma unroll
            for (int w = 1; w < HH; ++w) {
                run  *= C[w];
                tail *= C[w];
                E[g][w] = run;
            }
            E[g][0] = tail;
        }

        // LSTM cell update.
        float* orow = out + (t * BB + b) * HH;
        #pragma unroll
        for (int w = 0; w < HH; ++w) {
            const float f  = sigmoidf(E[0][w]);
            const float ii = sigmoidf(E[1][w]);
            const float gg = tanhf(E[2][w]);
            const float oo = sigmoidf(E[3][w]);
            cx[w] = f * cx[w] + ii * gg;
            hx[w] = oo * tanhf(cx[w]);
            orow[w] = hx[w];
        }
    }

    float* hxOut = out + (size_t)TT * BB * HH;
    float* cxOut = hxOut + (size_t)BB * HH;
    #pragma unroll
    for (int w = 0; w < HH; ++w) {
        hxOut[b * HH + w] = hx[w];
        cxOut[b * HH + w] = cx[w];
    }
}

// ---------------------------------------------------------------------------
extern "C" void kernel_launch(void* const* d_in, const int* in_sizes, int n_in,
                              void* d_out, int out_size, void* d_ws, size_t ws_size,
                              hipStream_t stream) {
    const float* x     = (const float*)d_in[0];  // (64,256,246)
    const float* W     = (const float*)d_in[1];  // (4,10,256)
    const float* bvec  = (const float*)d_in[2];  // (4,10)
    const float* theta = (const float*)d_in[3];  // (4,10)
    float* out = (float*)d_out;
    float* pre = (float*)d_ws;                   // 16384*40 floats = 2.62 MB

    // 1024 M-tiles * 3 N-tiles = 3072 waves; 4 waves per 128-thread block.
    qlstm_preact_wmma<<<768, 128, 0, stream>>>(x, W, bvec, theta, pre);
    // 256 independent batch lanes; 8 single-wave blocks spread across WGPs.
    qlstm_recurrence<<<8, 32, 0, stream>>>(pre, W, out);
}